// PostProcessor_38843684225287
// MI455X (gfx1250) — compile-verified
//
#include <hip/hip_runtime.h>

typedef __attribute__((ext_vector_type(4))) float v4f;

#define C_CLASSES 81
#define BBOX_XFORM_CLIP 4.135166556742356f   // log(1000/16)

// One wave32 per proposal:
//   lanes cooperatively softmax 81 logits (2 full strides + 17 tail),
//   decode the box redundantly (uniform data), then stream out
//   81 scores (b32, coalesced) and 81 box copies (b128, coalesced),
//   all with non-temporal hints (use-once streams, ~518 MB total traffic).
__global__ __launch_bounds__(256) void postproc_kernel(
    const float* __restrict__ logits,   // N*81
    const float* __restrict__ rel,      // N*4
    const float* __restrict__ prop,     // N*4
    const int*  __restrict__ img_w,
    const int*  __restrict__ img_h,
    float* __restrict__ boxes_out,      // N*81*4
    float* __restrict__ scores_out,     // N*81
    int N)
{
    const int lane = (int)(threadIdx.x & 31u);
    const int n    = (int)(blockIdx.x * (blockDim.x >> 5) + (threadIdx.x >> 5));
    if (n >= N) return;  // uniform per wave

    // ---------- wave-cooperative softmax over 81 logits ----------
    const float* row = logits + (size_t)n * C_CLASSES;
    float l0 = __builtin_nontemporal_load(row + lane);
    float l1 = __builtin_nontemporal_load(row + 32 + lane);
    float l2 = (lane < C_CLASSES - 64)
                   ? __builtin_nontemporal_load(row + 64 + lane)
                   : -__builtin_inff();

    float m = fmaxf(fmaxf(l0, l1), l2);
#pragma unroll
    for (int off = 16; off > 0; off >>= 1)
        m = fmaxf(m, __shfl_xor(m, off, 32));

    float e0 = __expf(l0 - m);
    float e1 = __expf(l1 - m);
    float e2 = (lane < C_CLASSES - 64) ? __expf(l2 - m) : 0.0f;

    float s = e0 + e1 + e2;
#pragma unroll
    for (int off = 16; off > 0; off >>= 1)
        s += __shfl_xor(s, off, 32);
    const float inv = 1.0f / s;

    // ---------- box decode (uniform across wave, computed per-lane) ----------
    const v4f rb = *(const v4f*)(rel  + (size_t)n * 4);  // 16B aligned
    const v4f pb = *(const v4f*)(prop + (size_t)n * 4);

    const float w  = pb.z - pb.x + 1.0f;
    const float h  = pb.w - pb.y + 1.0f;
    const float cx = pb.x + 0.5f * w;
    const float cy = pb.y + 0.5f * h;

    const float dx = rb.x / 10.0f;
    const float dy = rb.y / 10.0f;
    const float dw = fminf(rb.z / 5.0f, BBOX_XFORM_CLIP);
    const float dh = fminf(rb.w / 5.0f, BBOX_XFORM_CLIP);

    const float pcx = dx * w + cx;
    const float pcy = dy * h + cy;
    const float pw  = expf(dw) * w;
    const float ph  = expf(dh) * h;

    const float wmax = (float)(*img_w - 1);
    const float hmax = (float)(*img_h - 1);

    // max(.,0) then clip(0,max) collapses to a single clamp
    const float x1 = fminf(fmaxf(pcx - 0.5f * pw,        0.0f), wmax);
    const float y1 = fminf(fmaxf(pcy - 0.5f * ph,        0.0f), hmax);
    const float x2 = fminf(fmaxf(pcx + 0.5f * pw - 1.0f, 0.0f), wmax);
    const float y2 = fminf(fmaxf(pcy + 0.5f * ph - 1.0f, 0.0f), hmax);
    const v4f box = {x1, y1, x2, y2};

    // ---------- streaming stores (non-temporal) ----------
    float* srow = scores_out + (size_t)n * C_CLASSES;            // 4B aligned
    __builtin_nontemporal_store(e0 * inv, srow + lane);
    __builtin_nontemporal_store(e1 * inv, srow + 32 + lane);
    if (lane < C_CLASSES - 64)
        __builtin_nontemporal_store(e2 * inv, srow + 64 + lane);

    // boxes row base = n*1296 bytes, divisible by 16 -> b128 stores
    v4f* brow = (v4f*)(boxes_out + (size_t)n * (C_CLASSES * 4));
    __builtin_nontemporal_store(box, brow + lane);
    __builtin_nontemporal_store(box, brow + 32 + lane);
    if (lane < C_CLASSES - 64)
        __builtin_nontemporal_store(box, brow + 64 + lane);
}

extern "C" void kernel_launch(void* const* d_in, const int* in_sizes, int n_in,
                              void* d_out, int out_size, void* d_ws, size_t ws_size,
                              hipStream_t stream) {
    const float* logits = (const float*)d_in[0];   // class_logits  (N*81)
    const float* rel    = (const float*)d_in[1];   // box_regression (N*4)
    const float* prop   = (const float*)d_in[2];   // proposal_boxes (N*4)
    const int*  img_w   = (const int*)d_in[3];     // scalar
    const int*  img_h   = (const int*)d_in[4];     // scalar

    const int N = in_sizes[1] / 4;

    float* boxes  = (float*)d_out;                          // N*81*4 floats
    float* scores = boxes + (size_t)N * C_CLASSES * 4;      // N*81 floats

    const int wavesPerBlock = 8;                 // 256 threads = 8 wave32
    const int blocks = (N + wavesPerBlock - 1) / wavesPerBlock;
    postproc_kernel<<<blocks, 256, 0, stream>>>(logits, rel, prop, img_w, img_h,
                                                boxes, scores, N);
}